// LaneLoss_41420664602888
// MI455X (gfx1250) — compile-verified
//
#include <hip/hip_runtime.h>
#include <math.h>

typedef __attribute__((ext_vector_type(2))) float v2f;
typedef __attribute__((ext_vector_type(8))) float v8f;

#define BB 8
#define QQ 32
#define GG 16
#define NN 72
#define TT 7
#define NPAD 80  // 5 tiles of 16

// ---------------------------------------------------------------------------
// Kernel 1: stable compaction of gt points by visibility (visible first).
// One thread per (b,g); 128 threads total — trivial.
// ---------------------------------------------------------------------------
__global__ void lane_compact_kernel(const float* __restrict__ gt_pts,
                                    const float* __restrict__ gt_vis,
                                    float2* __restrict__ gtc,
                                    int* __restrict__ nvis) {
  int t = threadIdx.x;
  if (t >= BB * GG) return;
  const float* src = gt_pts + (size_t)t * NN * 2;
  const float* vis = gt_vis + (size_t)t * NN;
  float2* dst = gtc + (size_t)t * NN;
  int k = 0;
  for (int i = 0; i < NN; ++i) {
    if (vis[i] > 0.5f) { float2 v; v.x = src[2*i]; v.y = src[2*i+1]; dst[k++] = v; }
  }
  nvis[t] = k;
  for (int i = 0; i < NN; ++i) {
    if (!(vis[i] > 0.5f)) { float2 v; v.x = src[2*i]; v.y = src[2*i+1]; dst[k++] = v; }
  }
}

// ---------------------------------------------------------------------------
// Kernel 2: bidirectional chamfer geometry cost via V_WMMA_F32_16X16X4_F32.
// Block per (b,g); 8 waves, wave w handles q = w, w+8, w+16, w+24.
// For each direction: D1 = P(72x2,K pad 4) x AB(2x71) and D2 = P x A give the
// projection numerator and the p·a term; epilogue does clamp/min/sqrt.
// A frag:  lane%16 = row M, half-wave = K pair (K2,K3 zero-padded).
// B frag:  lane%16 = col N, half-wave = K pair.
// D frag:  lane%16 = col N, vgpr r + 8*(lane/16) = row M.
// tj loop fully unrolled: straight line of 10 WMMAs + epilogue per i-tile.
// ---------------------------------------------------------------------------
__global__ __launch_bounds__(256) void lane_cost_wmma_kernel(
    const float* __restrict__ pred,
    const float2* __restrict__ gtc,
    const int* __restrict__ nvisArr,
    float* __restrict__ geom) {
  int bg = blockIdx.x;
  int b = bg / GG, g = bg % GG;

  __shared__ float2 gpts[NPAD + 1];
  __shared__ float2 ppts[8][NPAD + 1];

  for (int i = threadIdx.x; i <= NPAD; i += 256) {
    float2 v; v.x = 0.f; v.y = 0.f;
    if (i < NN) v = gtc[(size_t)bg * NN + i];
    gpts[i] = v;
  }
  __syncthreads();

  int nv   = nvisArr[bg];
  int lane = threadIdx.x & 31;
  int wv   = threadIdx.x >> 5;
  int half = lane >> 4;   // 0 or 1
  int lj   = lane & 15;

  for (int qi = 0; qi < 4; ++qi) {
    int q = wv + 8 * qi;
    const float* pb = pred + (size_t)(b * QQ + q) * NN * 2;
    for (int i = lane; i <= NPAD; i += 32) {
      float2 v; v.x = 0.f; v.y = 0.f;
      if (i < NN) { v.x = pb[2*i]; v.y = pb[2*i+1]; }
      ppts[wv][i] = v;
    }
    __syncthreads();

    float dres[2];
    for (int dir = 0; dir < 2; ++dir) {
      const float2* Pp = dir ? gpts : ppts[wv];   // query points
      const float2* Lp = dir ? ppts[wv] : gpts;   // polyline
      int nSeg = dir ? (NN - 1) : (nv - 1);       // valid segs: j < nSeg
      int nPts = dir ? nv : NN;                   // weighted points: i < nPts

      float sum = 0.f;
      for (int ti = 0; ti < 5; ++ti) {
        int i0 = ti * 16;
        float2 pa = Pp[i0 + lj];
        v2f A;
        A.x = half ? 0.f : pa.x;  // K0 (lanes 0-15), K2=0 (lanes 16-31)
        A.y = half ? 0.f : pa.y;  // K1 / K3=0

        v8f acc;
        #pragma unroll
        for (int r = 0; r < 8; ++r) acc[r] = __builtin_inff();

        #pragma unroll
        for (int tj = 0; tj < 5; ++tj) {
          int j = tj * 16 + lj;
          float2 a  = Lp[j];
          float2 bp = Lp[j + 1];
          float abx = bp.x - a.x, aby = bp.y - a.y;
          float denom = fmaxf(abx * abx + aby * aby, 1e-9f);
          float adab  = a.x * abx + a.y * aby;
          float aa    = a.x * a.x + a.y * a.y;

          v2f Bf, B2f;
          Bf.x  = half ? 0.f : abx;  Bf.y  = half ? 0.f : aby;
          B2f.x = half ? 0.f : a.x;  B2f.y = half ? 0.f : a.y;

          v8f z = {0.f, 0.f, 0.f, 0.f, 0.f, 0.f, 0.f, 0.f};
          v8f D1 = __builtin_amdgcn_wmma_f32_16x16x4_f32(
              false, A, false, Bf, (short)0, z, false, false);   // p · ab
          v8f D2 = __builtin_amdgcn_wmma_f32_16x16x4_f32(
              false, A, false, B2f, (short)0, z, false, false);  // p · a

          float invd = __builtin_amdgcn_rcpf(denom);  // v_rcp_f32 (t is clamped)
          bool jok = (j < nSeg);
          #pragma unroll
          for (int r = 0; r < 8; ++r) {
            float num = D1[r] - adab;                       // ap·ab
            float t = fminf(fmaxf(num * invd, 0.f), 1.f);
            // d^2 - |p|^2  (|p|^2 is constant over j; added after min)
            float e = aa - 2.f * D2[r] + t * (t * denom - 2.f * num);
            acc[r] = fminf(acc[r], jok ? e : __builtin_inff());
          }
        }
        // min over the 16 columns held by each half-wave
        #pragma unroll
        for (int r = 0; r < 8; ++r) {
          float v = acc[r];
          v = fminf(v, __shfl_xor(v, 1));
          v = fminf(v, __shfl_xor(v, 2));
          v = fminf(v, __shfl_xor(v, 4));
          v = fminf(v, __shfl_xor(v, 8));
          acc[r] = v;
        }
        if (lj == 0) {
          #pragma unroll
          for (int r = 0; r < 8; ++r) {
            int i = i0 + half * 8 + r;
            if (i < nPts) {
              float2 p = Pp[i];
              float dper;
              if (nSeg > 0) {
                float pp = p.x * p.x + p.y * p.y;
                dper = __builtin_amdgcn_sqrtf(fmaxf(pp + acc[r], 0.f));
              } else {  // no valid segment: distance to poly[0]
                float dx = p.x - Lp[0].x, dy = p.y - Lp[0].y;
                dper = __builtin_amdgcn_sqrtf(dx * dx + dy * dy);
              }
              sum += dper;
            }
          }
        }
      }
      sum += __shfl_xor(sum, 16);       // lane0 += lane16
      sum = __shfl(sum, 0);
      float wsum = dir ? fmaxf((float)nv, 1.f) : (float)NN;
      dres[dir] = sum / wsum;
    }
    if (lane == 0) {
      float gval = (nv > 0) ? (dres[0] + dres[1]) : 1.0f;
      geom[(b * QQ + q) * GG + g] = gval;
    }
    __syncthreads();
  }
}

// ---------------------------------------------------------------------------
// Kernel 3: Hungarian (JV) assignment per batch, f64, one lane per batch.
// Mirrors the reference numpy implementation (rows = present gts, cols = Q).
// ---------------------------------------------------------------------------
__global__ void lane_hungarian_kernel(const float* __restrict__ geom,
                                      const float* __restrict__ exist,
                                      const float* __restrict__ gt_exist,
                                      const float* __restrict__ has_lanes,
                                      int* __restrict__ mcount,
                                      int* __restrict__ mq,
                                      int* __restrict__ mgi) {
  if (threadIdx.x != 0) return;
  int b = blockIdx.x;
  mcount[b] = 0;
  if (!(has_lanes[b] > 0.5f)) return;
  int idx[GG]; int n = 0;
  for (int g2 = 0; g2 < GG; ++g2)
    if (gt_exist[b * GG + g2] > 0.5f) idx[n++] = g2;
  if (n == 0) return;

  double u[GG + 1], v[QQ + 1];
  int p[QQ + 1], way[QQ + 1];
  for (int i = 0; i <= GG; ++i) u[i] = 0.0;
  for (int j = 0; j <= QQ; ++j) { v[j] = 0.0; p[j] = 0; way[j] = 0; }

  for (int i = 1; i <= n; ++i) {
    p[0] = i;
    int j0 = 0;
    double minv[QQ + 1]; bool used[QQ + 1];
    for (int j = 0; j <= QQ; ++j) { minv[j] = 1e300; used[j] = false; }
    while (true) {
      used[j0] = true;
      int i0 = p[j0];
      double delta = 1e300; int j1 = 0;
      for (int j = 1; j <= QQ; ++j) {
        if (!used[j]) {
          int qc = j - 1;
          float x = exist[b * QQ + qc];
          float sig = 1.f / (1.f + expf(-x));
          float cf = geom[(b * QQ + qc) * GG + idx[i0 - 1]] - 0.2f * sig;
          double cur = (double)cf - u[i0] - v[j];
          if (cur < minv[j]) { minv[j] = cur; way[j] = j0; }
          if (minv[j] < delta) { delta = minv[j]; j1 = j; }
        }
      }
      for (int j = 0; j <= QQ; ++j) {
        if (used[j]) { u[p[j]] += delta; v[j] -= delta; }
        else minv[j] -= delta;
      }
      j0 = j1;
      if (p[j0] == 0) break;
    }
    while (j0 != 0) { int j1 = way[j0]; p[j0] = p[j1]; j0 = j1; }
  }
  int ans[GG];
  for (int j = 1; j <= QQ; ++j) if (p[j] > 0) ans[p[j] - 1] = j - 1;
  for (int i = 0; i < n; ++i) {
    mq[b * GG + i]  = ans[i];
    mgi[b * GG + i] = idx[i];
  }
  mcount[b] = n;
}

// ---------------------------------------------------------------------------
// Kernel 4: final loss reduction (BCE + curve(reuse geom) + dir + smooth +
// type), single block of 256 threads, scalar output.
// ---------------------------------------------------------------------------
__global__ __launch_bounds__(256) void lane_loss_kernel(
    const float* __restrict__ exist, const float* __restrict__ pred,
    const float* __restrict__ tlog, const float* __restrict__ has_lanes,
    const int* __restrict__ gt_type, const float2* __restrict__ gtc,
    const int* __restrict__ nvisArr, const float* __restrict__ geom,
    const int* __restrict__ mcount, const int* __restrict__ mq,
    const int* __restrict__ mgi, float* __restrict__ out) {
  __shared__ float red[256];
  __shared__ int flag[BB * QQ];
  __shared__ int s_mb[BB * GG], s_mqv[BB * GG], s_mgv[BB * GG];
  __shared__ int s_M;

  int t = threadIdx.x;
  flag[t] = 0;
  if (t == 0) {
    int M = 0;
    for (int b2 = 0; b2 < BB; ++b2) {
      int c = mcount[b2];
      for (int i = 0; i < c; ++i) {
        s_mb[M] = b2; s_mqv[M] = mq[b2 * GG + i]; s_mgv[M] = mgi[b2 * GG + i];
        ++M;
      }
    }
    s_M = M;
  }
  __syncthreads();
  int M = s_M;
  if (t < M) flag[s_mb[t] * QQ + s_mqv[t]] = 1;
  __syncthreads();

  // BCE with POS_W = 3
  int b = t >> 5;
  float lm = has_lanes[b] > 0.5f ? 1.f : 0.f;
  float x = exist[t];
  float tt = (float)flag[t];
  float sp = fmaxf(-x, 0.f) + log1pf(expf(-fabsf(x)));  // softplus(-x)
  float bce = (1.f - tt) * x + (1.f + 2.f * tt) * sp;
  red[t] = bce * lm;
  __syncthreads();
  for (int s = 128; s > 0; s >>= 1) {
    if (t < s) red[t] += red[t + s];
    __syncthreads();
  }
  float bce_sum = red[0];
  __syncthreads();

  // per-match terms
  float val = 0.f;
  if (t < M) {
    int mb = s_mb[t], mqq = s_mqv[t], mgg = s_mgv[t];
    const float* pp = pred + (size_t)(mb * QQ + mqq) * NN * 2;
    const float2* gc = gtc + (size_t)(mb * GG + mgg) * NN;
    int nv = nvisArr[mb * GG + mgg];
    float curve = geom[(mb * QQ + mqq) * GG + mgg];

    float dsum = 0.f;
    for (int i = 0; i < NN - 1; ++i) {
      float dpx = pp[2*(i+1)]   - pp[2*i];
      float dpy = pp[2*(i+1)+1] - pp[2*i+1];
      float dgx = gc[i+1].x - gc[i].x;
      float dgy = gc[i+1].y - gc[i].y;
      float npn = fmaxf(sqrtf(dpx*dpx + dpy*dpy), 1e-6f);
      float ngn = fmaxf(sqrtf(dgx*dgx + dgy*dgy), 1e-6f);
      float c = (dpx*dgx + dpy*dgy) / (npn * ngn);
      if (i < nv - 1) dsum += 1.f - c;
    }
    int den = (nv - 1) > 1 ? (nv - 1) : 1;
    float dirt = (nv >= 3) ? (dsum / (float)den) : 0.f;

    float ss = 0.f;
    for (int i = 0; i < NN - 2; ++i) {
      float sx = pp[2*(i+2)]   - 2.f*pp[2*(i+1)]   + pp[2*i];
      float sy = pp[2*(i+2)+1] - 2.f*pp[2*(i+1)+1] + pp[2*i+1];
      ss += sqrtf(sx*sx + sy*sy);
    }
    float smooth = ss / (float)(NN - 2);

    const float* lg = tlog + (size_t)(mb * QQ + mqq) * TT;
    int cls = gt_type[mb * GG + mgg];
    float mx = lg[0];
    for (int c2 = 1; c2 < TT; ++c2) mx = fmaxf(mx, lg[c2]);
    float se = 0.f;
    for (int c2 = 0; c2 < TT; ++c2) se += expf(lg[c2] - mx);
    float typ = -(lg[cls] - (mx + logf(se)));

    val = 5.0f * curve + 1.5f * dirt + 0.5f * smooth + 1.0f * typ;
  }
  red[t] = val;
  __syncthreads();
  for (int s = 128; s > 0; s >>= 1) {
    if (t < s) red[t] += red[t + s];
    __syncthreads();
  }
  if (t == 0) {
    float lmsum = 0.f;
    for (int b2 = 0; b2 < BB; ++b2) lmsum += (has_lanes[b2] > 0.5f) ? 1.f : 0.f;
    float result = 0.f;
    if (lmsum > 0.f) {
      float exist_loss = bce_sum / (lmsum * (float)QQ);
      int nm = M > 0 ? M : 1;
      result = 2.0f * exist_loss + red[0] / (float)nm;
    }
    out[0] = result;
  }
}

// ---------------------------------------------------------------------------
extern "C" void kernel_launch(void* const* d_in, const int* in_sizes, int n_in,
                              void* d_out, int out_size, void* d_ws, size_t ws_size,
                              hipStream_t stream) {
  const float* exist   = (const float*)d_in[0];  // (B,Q,1)
  const float* pred    = (const float*)d_in[1];  // (B,Q,N,2)
  const float* tlog    = (const float*)d_in[2];  // (B,Q,T)
  const float* gtex    = (const float*)d_in[3];  // (B,G)
  const float* gtp     = (const float*)d_in[4];  // (B,G,N,2)
  const float* gtv     = (const float*)d_in[5];  // (B,G,N)
  const float* haslane = (const float*)d_in[6];  // (B,)
  const int*   gtt     = (const int*)d_in[7];    // (B,G) int32

  char* ws = (char*)d_ws;
  float2* gtc = (float2*)ws;                 //  73728 B
  int* nvis   = (int*)(ws + 73728);          //    512 B
  float* geom = (float*)(ws + 74240);        //  16384 B
  int* mcount = (int*)(ws + 90624);          //     32 B
  int* mq     = (int*)(ws + 90656);          //    512 B
  int* mgi    = (int*)(ws + 91168);          //    512 B
  float* out  = (float*)d_out;

  lane_compact_kernel<<<1, 128, 0, stream>>>(gtp, gtv, gtc, nvis);
  lane_cost_wmma_kernel<<<BB * GG, 256, 0, stream>>>(pred, gtc, nvis, geom);
  lane_hungarian_kernel<<<BB, 32, 0, stream>>>(geom, exist, gtex, haslane,
                                               mcount, mq, mgi);
  lane_loss_kernel<<<1, 256, 0, stream>>>(exist, pred, tlog, haslane, gtt,
                                          gtc, nvis, geom, mcount, mq, mgi, out);
}